// LSTMDecoder_73340861546583
// MI455X (gfx1250) — compile-verified
//
#include <hip/hip_runtime.h>
#include <hip/hip_bf16.h>
#include <math.h>
#include <float.h>

typedef float v2f __attribute__((ext_vector_type(2)));
typedef float v8f __attribute__((ext_vector_type(8)));

#define HID 256
#define BATCH 64

// ---- CDNA5 fp32 WMMA 16x16x4: D = A(16x4) * B(4x16) + C --------------------
__device__ __forceinline__ v8f wmma4(v2f a, v2f b, v8f c) {
  return __builtin_amdgcn_wmma_f32_16x16x4_f32(
      /*neg_a=*/false, a, /*neg_b=*/false, b,
      /*c_mod=*/(short)0, c, /*reuse_a=*/false, /*reuse_b=*/false);
}

__device__ __forceinline__ v2f ld2(const float* p) { return *(const v2f*)p; }
__device__ __forceinline__ float sigm(float x) { return 1.0f / (1.0f + expf(-x)); }

// ---- state init -------------------------------------------------------------
__global__ void k_state_init(int* tok, int* done, int* all_done, const int* __restrict__ sos) {
  int b = threadIdx.x;
  if (b < BATCH) { tok[b] = *sos; done[b] = 0; }
  if (b == 0) *all_done = 0;
}

// ---- h_init/c_init = enc @ W^T + b  (written to both layer-0 and layer-1 state)
__global__ void k_init_proj(const float* __restrict__ enc, const float* __restrict__ W,
                            const float* __restrict__ bias, float* __restrict__ OutA,
                            float* __restrict__ OutB, int F) {
  const int wave = threadIdx.x >> 5;
  const int lane = threadIdx.x & 31;
  const int m = lane & 15, hi = lane >> 4;
  const int koff = 2 * hi;
  const int rowbase = wave * 16;           // 4 waves -> 64 rows
  const int cbase = blockIdx.x * 16;       // 16 blocks -> 256 cols
  const int row = rowbase + m;
  const int col = cbase + m;
  const float* arow = enc + (size_t)row * F;
  const float* wrow = W + (size_t)col * F; // B[k][col] = W[col][k]
  v8f acc = {};
  for (int k = 0; k < F; k += 4) {
    acc = wmma4(ld2(arow + k + koff), ld2(wrow + k + koff), acc);
  }
  float bj = bias[col];
#pragma unroll
  for (int e = 0; e < 8; ++e) {
    int r = rowbase + e + 8 * hi;
    float v = acc[e] + bj;
    OutA[r * HID + col] = v;
    OutB[r * HID + col] = v;
  }
}

// ---- fused LSTM cell: gates = X@Wih^T + H@Whh^T + b; elementwise epilogue ---
// X is either emb (gathered via tok, K0=E) or a [64 x K0] state array (tok==null).
__global__ void k_lstm(const float* __restrict__ Xbase, const int* __restrict__ tok, int K0,
                       const float* __restrict__ Wih, const float* __restrict__ Whh,
                       const float* __restrict__ bih, const float* __restrict__ bhh,
                       const float* __restrict__ Hin, float* __restrict__ Cst,
                       float* __restrict__ Hout) {
  const int wave = threadIdx.x >> 5;
  const int lane = threadIdx.x & 31;
  const int m = lane & 15, hi = lane >> 4;
  const int koff = 2 * hi;
  const int rowbase = wave * 16;
  const int cbase = blockIdx.x * 16;       // 16 blocks cover H=256 columns
  const int row = rowbase + m;
  const int col = cbase + m;

  const float* arow = tok ? (Xbase + (size_t)tok[row] * K0)
                          : (Xbase + (size_t)row * K0);

  const float* w0 = Wih + (size_t)(0 * HID + col) * K0;  // gate i
  const float* w1 = Wih + (size_t)(1 * HID + col) * K0;  // gate f
  const float* w2 = Wih + (size_t)(2 * HID + col) * K0;  // gate g
  const float* w3 = Wih + (size_t)(3 * HID + col) * K0;  // gate o

  v8f acc0 = {}, acc1 = {}, acc2 = {}, acc3 = {};
  for (int k = 0; k < K0; k += 4) {
    v2f a = ld2(arow + k + koff);
    acc0 = wmma4(a, ld2(w0 + k + koff), acc0);
    acc1 = wmma4(a, ld2(w1 + k + koff), acc1);
    acc2 = wmma4(a, ld2(w2 + k + koff), acc2);
    acc3 = wmma4(a, ld2(w3 + k + koff), acc3);
  }
  const float* hrow = Hin + (size_t)row * HID;
  const float* u0 = Whh + (size_t)(0 * HID + col) * HID;
  const float* u1 = Whh + (size_t)(1 * HID + col) * HID;
  const float* u2 = Whh + (size_t)(2 * HID + col) * HID;
  const float* u3 = Whh + (size_t)(3 * HID + col) * HID;
  for (int k = 0; k < HID; k += 4) {
    v2f a = ld2(hrow + k + koff);
    acc0 = wmma4(a, ld2(u0 + k + koff), acc0);
    acc1 = wmma4(a, ld2(u1 + k + koff), acc1);
    acc2 = wmma4(a, ld2(u2 + k + koff), acc2);
    acc3 = wmma4(a, ld2(u3 + k + koff), acc3);
  }
  float bi = bih[0 * HID + col] + bhh[0 * HID + col];
  float bf = bih[1 * HID + col] + bhh[1 * HID + col];
  float bg = bih[2 * HID + col] + bhh[2 * HID + col];
  float bo = bih[3 * HID + col] + bhh[3 * HID + col];
#pragma unroll
  for (int e = 0; e < 8; ++e) {
    int r = rowbase + e + 8 * hi;
    float gi = sigm(acc0[e] + bi);
    float gf = sigm(acc1[e] + bf);
    float gg = tanhf(acc2[e] + bg);
    float go = sigm(acc3[e] + bo);
    float cold = Cst[r * HID + col];
    float cn = gf * cold + gi * gg;
    Cst[r * HID + col] = cn;
    Hout[r * HID + col] = go * tanhf(cn);
  }
}

// ---- logits chunk (64 cols) + per-row per-chunk argmax ----------------------
__global__ void k_logits(const float* __restrict__ H1, const float* __restrict__ Wout,
                         const float* __restrict__ bout, float* __restrict__ mval,
                         int* __restrict__ midx, int nchunks) {
  const int wave = threadIdx.x >> 5;
  const int lane = threadIdx.x & 31;
  const int m = lane & 15, hi = lane >> 4;
  const int koff = 2 * hi;
  const int rowbase = wave * 16;
  const int chunk = blockIdx.x;
  const int colbase = chunk * 64;
  const int row = rowbase + m;
  const float* arow = H1 + (size_t)row * HID;
  const float* w0 = Wout + (size_t)(colbase + 0 + m) * HID;
  const float* w1 = Wout + (size_t)(colbase + 16 + m) * HID;
  const float* w2 = Wout + (size_t)(colbase + 32 + m) * HID;
  const float* w3 = Wout + (size_t)(colbase + 48 + m) * HID;
  v8f acc0 = {}, acc1 = {}, acc2 = {}, acc3 = {};
  for (int k = 0; k < HID; k += 4) {
    v2f a = ld2(arow + k + koff);
    acc0 = wmma4(a, ld2(w0 + k + koff), acc0);
    acc1 = wmma4(a, ld2(w1 + k + koff), acc1);
    acc2 = wmma4(a, ld2(w2 + k + koff), acc2);
    acc3 = wmma4(a, ld2(w3 + k + koff), acc3);
  }
  const int c0 = colbase + m;
  float b0 = bout[c0], b1 = bout[c0 + 16], b2 = bout[c0 + 32], b3 = bout[c0 + 48];
#pragma unroll
  for (int e = 0; e < 8; ++e) {
    float bv = acc0[e] + b0; int bidx = c0;       // ascending cols, strict > keeps first max
    float v1 = acc1[e] + b1; if (v1 > bv) { bv = v1; bidx = c0 + 16; }
    float v2 = acc2[e] + b2; if (v2 > bv) { bv = v2; bidx = c0 + 32; }
    float v3 = acc3[e] + b3; if (v3 > bv) { bv = v3; bidx = c0 + 48; }
    // reduce across the 16 lanes of each half (rows differ between halves)
    for (int msk = 1; msk < 16; msk <<= 1) {
      float ov = __shfl_xor(bv, msk, 32);
      int oi = __shfl_xor(bidx, msk, 32);
      if (ov > bv || (ov == bv && oi < bidx)) { bv = ov; bidx = oi; }
    }
    if ((lane & 15) == 0) {
      int r = rowbase + e + 8 * hi;
      mval[(size_t)r * nchunks + chunk] = bv;
      midx[(size_t)r * nchunks + chunk] = bidx;
    }
  }
}

// ---- final argmax across chunks + token/done/all_done update ----------------
__global__ void k_select(const float* __restrict__ mval, const int* __restrict__ midx,
                         int nchunks, int* tok, int* done, int* all_done,
                         const int* __restrict__ eos, long long* __restrict__ out,
                         int t, int max_len) {
  __shared__ int cnt;
  int b = threadIdx.x;
  if (b == 0) cnt = 0;
  int ad = *all_done;  // previous-step value
  __syncthreads();
  float bv = -FLT_MAX; int bidx = 0;
  const float* vr = mval + (size_t)b * nchunks;
  const int* ir = midx + (size_t)b * nchunks;
  for (int c = 0; c < nchunks; ++c) {
    float v = vr[c];
    if (v > bv) { bv = v; bidx = ir[c]; }        // chunks ascending: first-max tie-break
  }
  int nt = ad ? 0 : bidx;
  out[(size_t)b * max_len + t] = (long long)nt;
  tok[b] = nt;
  int d = done[b] | (nt == *eos);
  done[b] = d;
  if (d) atomicAdd(&cnt, 1);
  __syncthreads();
  if (b == 0 && cnt == BATCH) *all_done = 1;
}

extern "C" void kernel_launch(void* const* d_in, const int* in_sizes, int n_in,
                              void* d_out, int out_size, void* d_ws, size_t ws_size,
                              hipStream_t stream) {
  (void)n_in; (void)ws_size;
  const float* enc  = (const float*)d_in[0];
  const float* emb  = (const float*)d_in[1];
  const float* Wh   = (const float*)d_in[2];
  const float* bh   = (const float*)d_in[3];
  const float* Wc   = (const float*)d_in[4];
  const float* bc   = (const float*)d_in[5];
  const float* Wih0 = (const float*)d_in[6];
  const float* Whh0 = (const float*)d_in[7];
  const float* bih0 = (const float*)d_in[8];
  const float* bhh0 = (const float*)d_in[9];
  const float* Wih1 = (const float*)d_in[10];
  const float* Whh1 = (const float*)d_in[11];
  const float* bih1 = (const float*)d_in[12];
  const float* bhh1 = (const float*)d_in[13];
  const float* Wout = (const float*)d_in[14];
  const float* bout = (const float*)d_in[15];
  const int* sos = (const int*)d_in[16];
  const int* eos = (const int*)d_in[17];

  const int V = in_sizes[15];            // 32000
  const int E = in_sizes[1] / V;         // 512
  const int F = in_sizes[2] / HID;       // 256
  const int ML = out_size / BATCH;       // 48
  const int nchunks = V / 64;            // 500 (V divisible by 64)

  // workspace layout (256B-aligned slabs)
  char* ws = (char*)d_ws;
  size_t off = 0;
  auto alloc = [&](size_t bytes) -> char* {
    char* p = ws + off;
    off = (off + bytes + 255) & ~(size_t)255;
    return p;
  };
  int* tok      = (int*)alloc(BATCH * 4);
  int* done     = (int*)alloc(BATCH * 4);
  int* all_done = (int*)alloc(4);
  float* h0[2]; h0[0] = (float*)alloc(BATCH * HID * 4); h0[1] = (float*)alloc(BATCH * HID * 4);
  float* c0     = (float*)alloc(BATCH * HID * 4);
  float* h1[2]; h1[0] = (float*)alloc(BATCH * HID * 4); h1[1] = (float*)alloc(BATCH * HID * 4);
  float* c1     = (float*)alloc(BATCH * HID * 4);
  float* mval   = (float*)alloc((size_t)BATCH * nchunks * 4);
  int* midx     = (int*)alloc((size_t)BATCH * nchunks * 4);

  long long* out = (long long*)d_out;
  dim3 blk(128);

  k_state_init<<<1, 64, 0, stream>>>(tok, done, all_done, sos);
  k_init_proj<<<HID / 16, blk, 0, stream>>>(enc, Wh, bh, h0[0], h1[0], F);
  k_init_proj<<<HID / 16, blk, 0, stream>>>(enc, Wc, bc, c0, c1, F);

  for (int t = 0; t < ML; ++t) {
    const int cur = t & 1, nxt = cur ^ 1;
    k_lstm<<<HID / 16, blk, 0, stream>>>(emb, tok, E, Wih0, Whh0, bih0, bhh0,
                                         h0[cur], c0, h0[nxt]);
    k_lstm<<<HID / 16, blk, 0, stream>>>(h0[nxt], nullptr, HID, Wih1, Whh1, bih1, bhh1,
                                         h1[cur], c1, h1[nxt]);
    k_logits<<<nchunks, blk, 0, stream>>>(h1[nxt], Wout, bout, mval, midx, nchunks);
    k_select<<<1, 64, 0, stream>>>(mval, midx, nchunks, tok, done, all_done, eos,
                                   out, t, ML);
  }
}